// LDS_2405181686067
// MI455X (gfx1250) — compile-verified
//
#include <hip/hip_runtime.h>

// ---------------------------------------------------------------------------
// Linear RNN on gfx1250, f32 WMMA throughout.
//   proj = x @ w_ih^T + b_ih                      (parallel GEMM, K=512)
//   v_t  = proj[:,t-1] @ w_hh^T + proj[:,t]       (parallel GEMM, K=1024)
//   h_t  = h_{t-2} @ (W^2)^T + v_t                (two parity chains, 1023
//                                                  sequential rounds each,
//                                                  run CONCURRENTLY)
// The sequential depth (device-wide barrier per round) dominates, so the
// parity split that halves it is the key algorithmic choice; the extra
// parallel 137-GFLOP GEMM is cheap against MI455X matrix throughput.
// Each scan WG builds its own 16-row W^2 slice in LDS via WMMA at startup
// (no global W^2 buffer; LDS ~72KB -> 4 WGs/WGP residency).
// This round: 4 independent split-K WMMA accumulator chains everywhere
// (RAW distance 4), compile-time K (no remainder loop), wave0-only fence.
// ---------------------------------------------------------------------------

typedef __attribute__((ext_vector_type(2))) float v2f;
typedef __attribute__((ext_vector_type(4))) float v4f;
typedef __attribute__((ext_vector_type(8))) float v8f;

#define Tn 2048
#define In 512
#define Hn 1024
#define Nb 32
#define NTH ((size_t)Nb * (size_t)Tn * (size_t)Hn)   // one output copy

#define GMAXK 1024
#define GSTR  (GMAXK + 4)     // LDS stride bound
#define RITER 16              // 32-row blocks per WG in the GEMM kernel

#define HSTR  (Hn + 4)        // padded stride for the W^2 slice
#define NWGC  64              // WGs per parity chain
#define NCH   2               // parity chains

#define WMMA_F32(a, b, c) \
  __builtin_amdgcn_wmma_f32_16x16x4_f32(false, (a), false, (b), (short)0, (c), false, false)

// ---------------------------------------------------------------------------
__global__ void init_bar_kernel(int* bar) {
  if (threadIdx.x < NCH) bar[threadIdx.x * 64] = 0;   // padded counters
}

// ---------------------------------------------------------------------------
// out[g,j] = sum_k A[g,k]*W[j,k] + (USEU ? U[g,j] : bias[j])
// grid = (Hn/32, M/(32*RITER)), block = 128.  Wave -> one 16x16 tile:
// waves {0,1} cols [0,16), waves {2,3} cols [16,32); wave&1 picks row half.
// 4 independent split-K WMMA chains per wave.
// ---------------------------------------------------------------------------
template <int K, int USEU>
__global__ __launch_bounds__(128) void gemm_kernel(
    const float* __restrict__ A, const float* __restrict__ W,
    const float* __restrict__ bias, const float* __restrict__ U,
    float* __restrict__ out)
{
  __shared__ float ws[32 * GSTR];
  __shared__ float xs[32 * GSTR];

  const int tid  = threadIdx.x;
  const int lane = tid & 31;
  const int wave = tid >> 5;
  const int j0   = blockIdx.x * 32;
  constexpr int kstr = K + 4;
  constexpr int kv   = K >> 2;                   // float4s per row

  // stage W[j0..j0+31][0..K) into LDS (float4, 4 threads per row)
  {
    const int r = tid >> 2;
    for (int c = (tid & 3); c < kv; c += 4)
      *(v4f*)&ws[r * kstr + c * 4] =
          *(const v4f*)&W[(size_t)(j0 + r) * K + c * 4];
  }

  const int lane15 = lane & 15;
  const int khalf  = (lane >> 4) << 1;
  const int nb     = (lane >> 4) ? 8 : 0;
  const int ct     = wave >> 1;                  // column tile
  const int mt     = wave & 1;                   // row half
  const int jcol   = j0 + ct * 16 + lane15;
  const float* wp  = &ws[(ct * 16 + lane15) * kstr + khalf];
  const float* xp  = &xs[(mt * 16 + lane15) * kstr + khalf];
  float biasv = 0.0f;
  if (!USEU) biasv = bias[jcol];

  for (int rb = 0; rb < RITER; ++rb) {
    const int m0 = (blockIdx.y * RITER + rb) * 32;
    __syncthreads();
    {
      const int r = tid >> 2;
      for (int c = (tid & 3); c < kv; c += 4)
        *(v4f*)&xs[r * kstr + c * 4] =
            *(const v4f*)&A[(size_t)(m0 + r) * K + c * 4];
    }
    __syncthreads();

    v8f acc0, acc1, acc2, acc3;
    if (USEU) {
      const float* up = U + (size_t)(m0 + mt * 16 + nb) * Hn + jcol;
#pragma unroll
      for (int r = 0; r < 8; ++r) acc0[r] = up[(size_t)r * Hn];
    } else {
#pragma unroll
      for (int r = 0; r < 8; ++r) acc0[r] = biasv;
    }
#pragma unroll
    for (int r = 0; r < 8; ++r) { acc1[r] = 0.f; acc2[r] = 0.f; acc3[r] = 0.f; }

#pragma unroll 2
    for (int k = 0; k < K; k += 16) {            // 4 independent chains
      v2f b0 = *(const v2f*)(wp + k);
      v2f a0 = *(const v2f*)(xp + k);
      acc0 = WMMA_F32(a0, b0, acc0);
      v2f b1 = *(const v2f*)(wp + k + 4);
      v2f a1 = *(const v2f*)(xp + k + 4);
      acc1 = WMMA_F32(a1, b1, acc1);
      v2f b2 = *(const v2f*)(wp + k + 8);
      v2f a2 = *(const v2f*)(xp + k + 8);
      acc2 = WMMA_F32(a2, b2, acc2);
      v2f b3 = *(const v2f*)(wp + k + 12);
      v2f a3 = *(const v2f*)(xp + k + 12);
      acc3 = WMMA_F32(a3, b3, acc3);
    }

    float* op = out + (size_t)(m0 + mt * 16 + nb) * Hn + jcol;
#pragma unroll
    for (int r = 0; r < 8; ++r)
      op[(size_t)r * Hn] = (acc0[r] + acc1[r]) + (acc2[r] + acc3[r]);
  }
}

// ---------------------------------------------------------------------------
// h0 fixup: copy1[:,0] = proj[:,0]  (overwrites the invalid t=0 rows the
// v-GEMM produced).  32*1024 elements.
// ---------------------------------------------------------------------------
__global__ void h0_kernel(float* __restrict__ out) {
  int idx = blockIdx.x * 256 + threadIdx.x;      // 128 blocks * 256
  int n = idx >> 10, j = idx & 1023;
  size_t o = (size_t)n * Tn * Hn + j;
  out[o] = out[NTH + o];
}

// ---------------------------------------------------------------------------
// Parity-chain scan.  grid = 128: blocks [0,64) even chain (t=2,4,...,2046),
// blocks [64,128) odd chain (t=3,5,...,2047).  Each WG owns 16 columns,
// builds its W^2 slice in LDS via WMMA, then runs 1023 rounds with an
// atomic grid barrier per parity group.
// ---------------------------------------------------------------------------
__global__ __launch_bounds__(128) void chain_kernel(
    const float* __restrict__ w_hh, float* __restrict__ out,
    int* __restrict__ bar)
{
  __shared__ float w2s[16 * HSTR];               // W^2 rows j0..j0+15
  __shared__ float red[3 * 2 * 8 * 32];          // wave partials

  const int tid  = threadIdx.x;
  const int lane = tid & 31;
  const int wave = tid >> 5;
  const int p    = blockIdx.x >> 6;              // parity
  const int wg   = blockIdx.x & 63;
  const int j0   = wg * 16;

  const int lane15 = lane & 15;
  const int khalf  = (lane >> 4) << 1;
  const int nb     = (lane >> 4) ? 8 : 0;

  // ---- build W^2[j0..j0+15][:] = (W @ W) slice, via WMMA --------------------
  // D[j',k'] = sum_m W[j0+j', m] * W[m, k0+k']
  {
    const float* ap = w_hh + (size_t)(j0 + lane15) * Hn + khalf; // A rows
    for (int kt = wave * 16; kt < wave * 16 + 16; ++kt) {
      const int k0 = kt * 16;
      const float* bcol = w_hh + (size_t)khalf * Hn + k0 + lane15;
      v8f acc = {}, accB = {};
#pragma unroll 2
      for (int m = 0; m < Hn; m += 8) {
        v2f a0 = *(const v2f*)(ap + m);
        v2f b0;
        b0.x = bcol[(size_t)m * Hn];
        b0.y = bcol[(size_t)(m + 1) * Hn];
        acc  = WMMA_F32(a0, b0, acc);
        v2f a1 = *(const v2f*)(ap + m + 4);
        v2f b1;
        b1.x = bcol[(size_t)(m + 4) * Hn];
        b1.y = bcol[(size_t)(m + 5) * Hn];
        accB = WMMA_F32(a1, b1, accB);
      }
#pragma unroll
      for (int r = 0; r < 8; ++r)
        w2s[(nb + r) * HSTR + k0 + lane15] = acc[r] + accB[r];
    }
  }
  __syncthreads();

  const int jcol    = j0 + lane15;
  const int kbase   = wave * 256;                // K split across 4 waves
  const float* wpk  = &w2s[lane15 * HSTR + khalf] + kbase;

  for (int r2 = 1; r2 < Tn / 2; ++r2) {          // 1023 rounds
    const int t = 2 * r2 + p;
    // h_{t-2} rows: copy1[(n*T + t-2)*H + k]
    const float* a0base = out + ((size_t)lane15 * Tn + (size_t)(t - 2)) * Hn
                              + kbase + khalf;
    const float* a1base = a0base + (size_t)16 * Tn * Hn;

    v8f acc0a = {}, acc0b = {}, acc1a = {}, acc1b = {};
#pragma unroll 4
    for (int kk = 0; kk < 256; kk += 8) {        // 4 independent chains
      v2f a0 = *(const v2f*)(a0base + kk);
      v2f a1 = *(const v2f*)(a1base + kk);
      v2f b0 = *(const v2f*)(wpk + kk);
      acc0a = WMMA_F32(a0, b0, acc0a);
      acc1a = WMMA_F32(a1, b0, acc1a);
      v2f a2 = *(const v2f*)(a0base + kk + 4);
      v2f a3 = *(const v2f*)(a1base + kk + 4);
      v2f b1 = *(const v2f*)(wpk + kk + 4);
      acc0b = WMMA_F32(a2, b1, acc0b);
      acc1b = WMMA_F32(a3, b1, acc1b);
    }

    v8f acc0, acc1;
#pragma unroll
    for (int r = 0; r < 8; ++r) {
      acc0[r] = acc0a[r] + acc0b[r];
      acc1[r] = acc1a[r] + acc1b[r];
    }

    if (wave != 0) {
      float* rp = &red[(wave - 1) * 512];
#pragma unroll
      for (int r = 0; r < 8; ++r) {
        rp[r * 32 + lane]       = acc0[r];
        rp[256 + r * 32 + lane] = acc1[r];
      }
    }
    __syncthreads();

    if (wave == 0) {
#pragma unroll
      for (int w2 = 0; w2 < 3; ++w2) {
        const float* rp = &red[w2 * 512];
#pragma unroll
        for (int r = 0; r < 8; ++r) {
          acc0[r] += rp[r * 32 + lane];
          acc1[r] += rp[256 + r * 32 + lane];
        }
      }
      // h_t = partial + v_t (v already sitting in copy1[:,t]); store copy1 only
#pragma unroll
      for (int r = 0; r < 8; ++r) {
        size_t i0 = ((size_t)(nb + r) * Tn + (size_t)t) * Hn + jcol;
        size_t i1 = ((size_t)(nb + r + 16) * Tn + (size_t)t) * Hn + jcol;
        out[i0] = acc0[r] + out[i0];
        out[i1] = acc1[r] + out[i1];
      }
      __threadfence();                           // release wave0's h stores
    }
    __syncthreads();                             // red consumed, stores done

    if (tid == 0) {
      int* mybar = bar + p * 64;                 // padded counter per parity
      __hip_atomic_fetch_add(mybar, 1, __ATOMIC_ACQ_REL,
                             __HIP_MEMORY_SCOPE_AGENT);
      const int target = NWGC * r2;
      while (__hip_atomic_load(mybar, __ATOMIC_ACQUIRE,
                               __HIP_MEMORY_SCOPE_AGENT) < target) {
        __builtin_amdgcn_s_sleep(1);
      }
    }
    __syncthreads();
  }
}

// ---------------------------------------------------------------------------
// Final duplication: copy2 = copy1 (pure bandwidth, ~23us at 23.3 TB/s).
// ---------------------------------------------------------------------------
__global__ void dup_kernel(float* __restrict__ out) {
  const size_t nvec = NTH / 4;
  size_t idx = (size_t)blockIdx.x * 256 + threadIdx.x;
  const size_t stride = (size_t)gridDim.x * 256;
  v4f* dst = (v4f*)(out + NTH);
  const v4f* src = (const v4f*)out;
  for (; idx < nvec; idx += stride) dst[idx] = src[idx];
}

// ---------------------------------------------------------------------------
extern "C" void kernel_launch(void* const* d_in, const int* in_sizes, int n_in,
                              void* d_out, int out_size, void* d_ws, size_t ws_size,
                              hipStream_t stream) {
  const float* x    = (const float*)d_in[0];
  const float* w_ih = (const float*)d_in[1];
  const float* b_ih = (const float*)d_in[2];
  const float* w_hh = (const float*)d_in[3];
  float* out  = (float*)d_out;
  float* out2 = out + NTH;                       // copy-2 region = scratch
  int*   bar  = (int*)d_ws;

  init_bar_kernel<<<1, 32, 0, stream>>>(bar);

  dim3 gg(Hn / 32, (Nb * Tn) / (32 * RITER));    // (32, 128)

  // proj = x @ w_ih^T + b_ih  -> copy-2 region (scratch)
  gemm_kernel<In, 0><<<gg, 128, 0, stream>>>(x, w_ih, b_ih, nullptr, out2);

  // v_g = proj[g-1] @ w_hh^T + proj[g]  -> copy-1 (t=0 rows fixed below)
  gemm_kernel<Hn, 1><<<gg, 128, 0, stream>>>(out2 - Hn, w_hh, nullptr, out2, out);

  // h_0 = proj[:,0]
  h0_kernel<<<128, 256, 0, stream>>>(out);

  // two concurrent parity chains, 1023 barrier rounds each
  chain_kernel<<<NCH * NWGC, 128, 0, stream>>>(w_hh, out, bar);

  // duplicate output
  dup_kernel<<<2048, 256, 0, stream>>>(out);
}